// NAOALoss_18107582120055
// MI455X (gfx1250) — compile-verified
//
#include <hip/hip_runtime.h>
#include <math.h>

#define N_PTS 8192
#define TILE_M 16
#define CHUNK 1024
#define WAVES_PER_BLOCK 8
#define THREADS (WAVES_PER_BLOCK * 32)
#define NUM_TILES (N_PTS / TILE_M) /* 512 */
#define K_RADIUS_F 2.0f
#define LOSS_WEIGHT_F 1.0f

typedef __attribute__((ext_vector_type(2))) float v2f;
typedef __attribute__((ext_vector_type(8))) float v8f;

// ---------------------------------------------------------------------------
// Kernel 1: per-point precompute
//   cx, cy            : centers (stop-grad, fwd only)
//   inv2sig           : 1 / (2 * (K_RADIUS * clamp(sqrt(w*h),16,800))^2)
//   vx, vy            : cos(4*theta), sin(4*theta)
//   sc                : score^SCORE_ALPHA (alpha = 1 -> score)
//   key               : batch*16 + label  (equality <=> both masks true)
// ---------------------------------------------------------------------------
__global__ void naoa_precompute(const float* __restrict__ bbox,
                                const float* __restrict__ scores,
                                const int* __restrict__ labels,
                                const int* __restrict__ batch,
                                float* __restrict__ cx, float* __restrict__ cy,
                                float* __restrict__ vx, float* __restrict__ vy,
                                float* __restrict__ sc, float* __restrict__ inv2sig,
                                int* __restrict__ key) {
  int i = blockIdx.x * blockDim.x + threadIdx.x;
  if (i >= N_PTS) return;
  float x  = bbox[i * 5 + 0];
  float y  = bbox[i * 5 + 1];
  float w  = bbox[i * 5 + 2];
  float h  = bbox[i * 5 + 3];
  float th = bbox[i * 5 + 4];
  float scale = sqrtf(w * h);
  scale = fminf(fmaxf(scale, 16.0f), 800.0f);
  float sig = scale * K_RADIUS_F;
  cx[i] = x;
  cy[i] = y;
  inv2sig[i] = 1.0f / (2.0f * sig * sig);
  float s, c;
  __sincosf(4.0f * th, &s, &c);
  vx[i] = c;
  vy[i] = s;
  sc[i] = scores[i];
  key[i] = batch[i] * 16 + labels[i];
}

// ---------------------------------------------------------------------------
// Kernel 2: pairwise tile kernel.
// One wave per 16-row i-tile. j-data staged through LDS in CHUNK entries.
// Per inner iteration: 4 j's -> one v_wmma_f32_16x16x4_f32 accumulating
//   D[i,0] += W[i,j],  D[i,1] += W[i,j]*vx_j,  D[i,2] += W[i,j]*vy_j
// A layout (f32 16x4): lane holds row i = lane%16; VGPR0/1 = K {0,1} for
// lanes 0-15 and K {2,3} for lanes 16-31. B layout mirrors (row K striped).
// ---------------------------------------------------------------------------
__global__ void __launch_bounds__(THREADS)
naoa_pairwise(const float* __restrict__ cx, const float* __restrict__ cy,
              const float* __restrict__ vx, const float* __restrict__ vy,
              const float* __restrict__ sc, const float* __restrict__ inv2sig,
              const int* __restrict__ key, float* __restrict__ partials) {
  __shared__ float4 jgeo[CHUNK];                       // cx, cy, vx, vy
  __shared__ float2 jsk[CHUNK];                        // score, key bits
  __shared__ float stage[WAVES_PER_BLOCK][TILE_M][3];  // Sw, Sx, Sy per row

  const int lane = threadIdx.x & 31;
  const int wave = threadIdx.x >> 5;
  const int tile = blockIdx.x * WAVES_PER_BLOCK + wave;
  const int n = lane & 15;            // column in 16x16 tile / row in i-tile
  const int klo = (lane >> 4) << 1;   // K offset: 0 (lanes 0-15), 2 (16-31)
  const int row = tile * TILE_M + n;

  const float cxi  = cx[row];
  const float cyi  = cy[row];
  const float inv2 = inv2sig[row];
  const int   keyi = key[row];

  v8f acc = {0.0f, 0.0f, 0.0f, 0.0f, 0.0f, 0.0f, 0.0f, 0.0f};

  for (int base = 0; base < N_PTS; base += CHUNK) {
    __syncthreads();  // protect LDS reuse across chunks
    for (int t = threadIdx.x; t < CHUNK; t += THREADS) {
      int j = base + t;
      jgeo[t] = make_float4(cx[j], cy[j], vx[j], vy[j]);
      jsk[t]  = make_float2(sc[j], __int_as_float(key[j]));
    }
    __syncthreads();

    for (int jj = 0; jj < CHUNK; jj += 4) {
      const int jA = jj + klo;
      float4 gA = jgeo[jA];
      float2 kA = jsk[jA];
      float4 gB = jgeo[jA + 1];
      float2 kB = jsk[jA + 1];

      float dxA = cxi - gA.x, dyA = cyi - gA.y;
      float mA  = (__float_as_int(kA.y) == keyi) ? kA.x : 0.0f;
      float wA  = __expf(-(dxA * dxA + dyA * dyA) * inv2) * mA;

      float dxB = cxi - gB.x, dyB = cyi - gB.y;
      float mB  = (__float_as_int(kB.y) == keyi) ? kB.x : 0.0f;
      float wB  = __expf(-(dxB * dxB + dyB * dyB) * inv2) * mB;

      v2f a;
      a.x = wA;
      a.y = wB;
      // B columns: 0 -> ones (row sum), 1 -> vx, 2 -> vy, 3..15 -> 0
      float bA = (n == 0) ? 1.0f : (n == 1) ? gA.z : (n == 2) ? gA.w : 0.0f;
      float bB = (n == 0) ? 1.0f : (n == 1) ? gB.z : (n == 2) ? gB.w : 0.0f;
      v2f b;
      b.x = bA;
      b.y = bB;

      acc = __builtin_amdgcn_wmma_f32_16x16x4_f32(
          /*neg_a=*/false, a, /*neg_b=*/false, b,
          /*c_mod=*/(short)0, acc, /*reuse_a=*/false, /*reuse_b=*/false);
    }
  }

  // Extract columns 0..2 of C/D layout: lane holds N=lane%16,
  // VGPR v holds M = v + 8*(lane>=16).
  float accA[8];
  *(v8f*)accA = acc;
  if (n < 3) {
    const int mbase = (lane >> 4) * 8;
#pragma unroll
    for (int v = 0; v < 8; ++v)
      stage[wave][mbase + v][n] = accA[v];
  }
  __syncthreads();

  float sw = stage[wave][n][0];
  float sx = stage[wave][n][1];
  float sy = stage[wave][n][2];
  // mean_vec = (sx/sw, sy/sw); chaos = 1 - |mean_vec|. sw > 0 (diagonal term).
  float chaos = 1.0f - sqrtf(sx * sx + sy * sy) / sw;
  float val = (lane < 16) ? chaos : 0.0f;
#pragma unroll
  for (int off = 8; off >= 1; off >>= 1)
    val += __shfl_xor(val, off, 32);
  if (lane == 0) partials[tile] = val;
}

// ---------------------------------------------------------------------------
// Kernel 3: deterministic fixed-order reduction of 512 tile partials.
// ---------------------------------------------------------------------------
__global__ void naoa_finalize(const float* __restrict__ partials,
                              float* __restrict__ out) {
  __shared__ float red[NUM_TILES];
  int t = threadIdx.x;
  red[t] = partials[t];
  __syncthreads();
  for (int s = NUM_TILES / 2; s >= 1; s >>= 1) {
    if (t < s) red[t] += red[t + s];
    __syncthreads();
  }
  if (t == 0) out[0] = LOSS_WEIGHT_F * red[0] / (float)N_PTS;
}

// ---------------------------------------------------------------------------
extern "C" void kernel_launch(void* const* d_in, const int* in_sizes, int n_in,
                              void* d_out, int out_size, void* d_ws, size_t ws_size,
                              hipStream_t stream) {
  const float* bbox   = (const float*)d_in[0];  // N x 5
  const float* scores = (const float*)d_in[1];  // N
  const int*   labels = (const int*)d_in[2];    // N
  const int*   batch  = (const int*)d_in[3];    // N

  float* f        = (float*)d_ws;
  float* cx       = f + 0 * N_PTS;
  float* cy       = f + 1 * N_PTS;
  float* vx       = f + 2 * N_PTS;
  float* vy       = f + 3 * N_PTS;
  float* sc       = f + 4 * N_PTS;
  float* inv2sig  = f + 5 * N_PTS;
  int*   key      = (int*)(f + 6 * N_PTS);
  float* partials = f + 7 * N_PTS;  // NUM_TILES floats

  naoa_precompute<<<N_PTS / 256, 256, 0, stream>>>(
      bbox, scores, labels, batch, cx, cy, vx, vy, sc, inv2sig, key);

  naoa_pairwise<<<NUM_TILES / WAVES_PER_BLOCK, THREADS, 0, stream>>>(
      cx, cy, vx, vy, sc, inv2sig, key, partials);

  naoa_finalize<<<1, NUM_TILES, 0, stream>>>(partials, (float*)d_out);
}